// CrossAttention_37555194036871
// MI455X (gfx1250) — compile-verified
//
#include <hip/hip_runtime.h>

// ---------------------------------------------------------------------------
// Cross-attention forward, fp32 end-to-end, CDNA5 (gfx1250, wave32).
//
// Roofline: ~103 GFLOP vs ~60 MB mandatory HBM traffic -> memory-bound at
// 23.3 TB/s even on the fp32 matrix pipe (V_WMMA_F32_16X16X4_F32), so we keep
// full fp32 accuracy and optimize data movement:
//   * flash-style fused attention (never materialize B*H*S*S scores)
//   * row-major LDS tiles so every WMMA fragment is one aligned ds_load_b64
//   * async global->LDS staging (GLOBAL_LOAD_ASYNC_TO_LDS_B128 + ASYNCcnt)
//     with a double-buffered single-barrier pipeline in the GEMMs
//   * L2 prefetch of the next K/V tile during attention compute
//
// Head reshape note: reference reshapes (B,S,1024)->(B,16,2048,64) with NO
// transpose, so head h of batch b is the contiguous block at element offset
// (b*16+h)*2048*64, viewed as row-major 2048x64.
// ---------------------------------------------------------------------------

typedef float v2f __attribute__((ext_vector_type(2)));
typedef float v4f __attribute__((ext_vector_type(4)));
typedef float v8f __attribute__((ext_vector_type(8)));
typedef int   v4i __attribute__((ext_vector_type(4)));

#define WMMA_F32(a, b, c) \
  __builtin_amdgcn_wmma_f32_16x16x4_f32(false, (a), false, (b), (short)0, (c), false, false)

#ifndef __has_builtin
#define __has_builtin(x) 0
#endif
#if __has_builtin(__builtin_amdgcn_global_load_async_to_lds_b128) && \
    __has_builtin(__builtin_amdgcn_s_wait_asynccnt)
#define USE_ASYNC_COPY 1
// builtin signature (from the round-2 diagnostic): param0 = v4i AS1* (global),
// param1 = v4i AS3* (LDS), then imm offset, imm cpol.
typedef __attribute__((address_space(1))) v4i* async_g_t;
typedef __attribute__((address_space(3))) v4i* async_l_t;
// per-lane 16B copy: global address srcGlb to LDS address dstLds, tracked by ASYNCcnt
#define ASYNC_CP16(dstLds, srcGlb) \
  __builtin_amdgcn_global_load_async_to_lds_b128((async_g_t)(srcGlb), (async_l_t)(dstLds), 0, 0)
#define ASYNC_WAIT() __builtin_amdgcn_s_wait_asynccnt(0)
#else
#define USE_ASYNC_COPY 0
#define ASYNC_WAIT()
#endif

constexpr int   kB    = 2;
constexpr int   kS    = 2048;
constexpr int   kD    = 1024;
constexpr int   kH    = 16;
constexpr int   kDH   = 64;       // head dim of the (faithful-to-bug) reshape view
constexpr float kNorm = 0.03125f; // 1/sqrt(1024)

// ===========================================================================
// GEMM: Y[M,N] = X[M,K] @ W[N,K]^T + bias[N]   (torch-style weight layout)
// Block tile 128x128, K-tile 16, 256 threads (8 waves, 2x4), wave tile 64x32.
// Tiles staged ROW-MAJOR (As[m][k], Bs[n][k], pad 4 -> 80B rows):
//   * staging is a pure per-lane 16B copy -> async-to-LDS, double buffered
//   * A/B fragments are single aligned ds_load_b64 (adjacent k, k+1)
// ===========================================================================
#define GBM 128
#define GBN 128
#define GBK 16
#define GPAD 4

__global__ __launch_bounds__(256)
void gemm_bias_wmma(const float* __restrict__ X, const float* __restrict__ W,
                    const float* __restrict__ bias, float* __restrict__ Y,
                    int M, int N, int Kd) {
  __shared__ float As[2][GBM][GBK + GPAD];   // X tile, row-major
  __shared__ float Bs[2][GBN][GBK + GPAD];   // W tile, row-major (B[k][n] = Bs[n][k])

  const int tid  = threadIdx.x;
  const int lane = tid & 31;
  const int lo   = lane & 15;
  const int hi   = lane >> 4;
  const int w    = tid >> 5;
  const int wm   = (w & 1) * 64;
  const int wn   = (w >> 1) * 32;
  const int bm   = blockIdx.y * GBM;
  const int bn   = blockIdx.x * GBN;

  // 16B-chunk staging map: 128 rows x 16 floats = 512 chunks, 2 per thread
  auto stage = [&](int kt, int buf) {
    const int k0 = kt * GBK;
#pragma unroll
    for (int p = 0; p < 2; ++p) {
      const int idx = tid + p * 256;
      const int row = idx >> 2;
      const int c4  = (idx & 3) * 4;
#if USE_ASYNC_COPY
      ASYNC_CP16(&As[buf][row][c4], X + (size_t)(bm + row) * Kd + k0 + c4);
      ASYNC_CP16(&Bs[buf][row][c4], W + (size_t)(bn + row) * Kd + k0 + c4);
#else
      *(v4f*)&As[buf][row][c4] = *(const v4f*)(X + (size_t)(bm + row) * Kd + k0 + c4);
      *(v4f*)&Bs[buf][row][c4] = *(const v4f*)(W + (size_t)(bn + row) * Kd + k0 + c4);
#endif
    }
  };

  v8f acc[4][2];
#pragma unroll
  for (int mt = 0; mt < 4; ++mt)
#pragma unroll
    for (int nt = 0; nt < 2; ++nt)
#pragma unroll
      for (int r = 0; r < 8; ++r) acc[mt][nt][r] = 0.0f;

  const int NT = Kd / GBK;
  stage(0, 0);
  ASYNC_WAIT();
  __syncthreads();

  for (int kt = 0; kt < NT; ++kt) {
    const int buf = kt & 1;
    // issue async copies for the next tile; they stream while we do WMMAs
    if (kt + 1 < NT) stage(kt + 1, buf ^ 1);

#pragma unroll
    for (int ks = 0; ks < 4; ++ks) {
      const int k0 = ks * 4 + 2 * hi;   // A-layout: lanes 16-31 take k0+2,k0+3
      v2f af[4], bf[2];
#pragma unroll
      for (int mt = 0; mt < 4; ++mt)
        af[mt] = *(const v2f*)&As[buf][wm + mt * 16 + lo][k0];
#pragma unroll
      for (int nt = 0; nt < 2; ++nt)
        bf[nt] = *(const v2f*)&Bs[buf][wn + nt * 16 + lo][k0];
#pragma unroll
      for (int mt = 0; mt < 4; ++mt)
#pragma unroll
        for (int nt = 0; nt < 2; ++nt)
          acc[mt][nt] = WMMA_F32(af[mt], bf[nt], acc[mt][nt]);
    }

    ASYNC_WAIT();        // next tile resident
    __syncthreads();     // all waves done reading current tile
  }

  // epilogue: C layout row = r + 8*hi, col = lo
#pragma unroll
  for (int mt = 0; mt < 4; ++mt) {
#pragma unroll
    for (int nt = 0; nt < 2; ++nt) {
      const int   gn = bn + wn + nt * 16 + lo;
      const float bv = bias[gn];
#pragma unroll
      for (int r = 0; r < 8; ++r) {
        const int gm = bm + wm + mt * 16 + r + 8 * hi;
        Y[(size_t)gm * N + gn] = acc[mt][nt][r] + bv;
      }
    }
  }
}

// ===========================================================================
// Fused flash attention per (b, h): Q,K,V contiguous row-major 2048x64.
// 128 threads = 4 waves; block owns 64 q-rows (16 per wave), iterates over
// 32 k-blocks of 64. K/V tiles async-staged to LDS; next tile L2-prefetched
// during compute. S = (Q*norm)*K^T via WMMA, online softmax over 16-lane
// half-groups, P bounced through per-wave LDS to re-enter WMMA as A-frags.
// ===========================================================================
__global__ __launch_bounds__(128)
void flash_attn_wmma(const float* __restrict__ Qb, const float* __restrict__ Kb,
                     const float* __restrict__ Vb, const float* __restrict__ Mask,
                     float* __restrict__ Ob) {
  __shared__ float Ks[64][kDH + 4];      // K tile [k-row][dh], 272B rows
  __shared__ float Vs[64][kDH + 4];      // V tile [k-row][dh]
  __shared__ float Ps[4][16][kDH + 4];   // per-wave P bounce (C->A layout)

  const int tid  = threadIdx.x;
  const int lane = tid & 31;
  const int lo   = lane & 15;
  const int hi   = lane >> 4;
  const int wv   = tid >> 5;

  const size_t headOff = ((size_t)blockIdx.z * kH + blockIdx.y) * (size_t)kS * kDH;
  const float* Q = Qb + headOff;
  const float* K = Kb + headOff;
  const float* V = Vb + headOff;
  float*       O = Ob + headOff;

  const int q0 = blockIdx.x * 64 + wv * 16;

  // Q as A-fragments for all 16 k-steps (dh=64); adjacent pair -> b64 load
  v2f qa[16];
  {
    const float* qrow = Q + (size_t)(q0 + lo) * kDH;
#pragma unroll
    for (int ks = 0; ks < 16; ++ks) {
      v2f q2 = *(const v2f*)(qrow + ks * 4 + 2 * hi);
      qa[ks].x = q2.x * kNorm;
      qa[ks].y = q2.y * kNorm;
    }
  }

  v8f   o[4];
  float mrow[8], lrow[8];
#pragma unroll
  for (int nt = 0; nt < 4; ++nt)
#pragma unroll
    for (int r = 0; r < 8; ++r) o[nt][r] = 0.0f;
#pragma unroll
  for (int r = 0; r < 8; ++r) { mrow[r] = -__builtin_inff(); lrow[r] = 0.0f; }

  for (int kb = 0; kb < kS / 64; ++kb) {
    // stage K,V tiles: 64x64 floats = 1024 16B chunks, 8 per thread
#pragma unroll
    for (int i = 0; i < 8; ++i) {
      const int idx = tid + i * 128;
      const int row = idx >> 4;
      const int c4  = (idx & 15) * 4;
      const size_t g = (size_t)(kb * 64 + row) * kDH + c4;
#if USE_ASYNC_COPY
      ASYNC_CP16(&Ks[row][c4], K + g);
      ASYNC_CP16(&Vs[row][c4], V + g);
#else
      *(v4f*)&Ks[row][c4] = *(const v4f*)(K + g);
      *(v4f*)&Vs[row][c4] = *(const v4f*)(V + g);
#endif
    }
    // L2 prefetch of next tile (1 cacheline per thread per tensor)
    if (kb + 1 < kS / 64) {
      const size_t pf = (size_t)(kb + 1) * 64 * kDH + (size_t)tid * 32;
      __builtin_prefetch((const void*)(K + pf), 0, 1);
      __builtin_prefetch((const void*)(V + pf), 0, 1);
    }
    ASYNC_WAIT();
    __syncthreads();

    // S tile: 16 (q) x 64 (k) = 4 WMMA C tiles; B[k][n] = Ks[n][k] (adjacent)
    v8f st[4];
#pragma unroll
    for (int nt = 0; nt < 4; ++nt) {
      v8f s = {0.f, 0.f, 0.f, 0.f, 0.f, 0.f, 0.f, 0.f};
#pragma unroll
      for (int ks = 0; ks < 16; ++ks) {
        v2f bf = *(const v2f*)&Ks[nt * 16 + lo][ks * 4 + 2 * hi];
        s = WMMA_F32(qa[ks], bf, s);
      }
      st[nt] = s;
    }

    // + attn_mask (broadcast over b,h; zeros here but faithful)
    {
      const float* mbase = Mask + (size_t)q0 * kS + (size_t)kb * 64;
#pragma unroll
      for (int nt = 0; nt < 4; ++nt)
#pragma unroll
        for (int r = 0; r < 8; ++r)
          st[nt][r] += mbase[(size_t)(r + 8 * hi) * kS + nt * 16 + lo];
    }

    // online softmax: per row (r + 8*hi), reduce over 4 tiles x 16 lanes
#pragma unroll
    for (int r = 0; r < 8; ++r) {
      float mx = fmaxf(fmaxf(st[0][r], st[1][r]), fmaxf(st[2][r], st[3][r]));
#pragma unroll
      for (int off = 1; off <= 8; off <<= 1)
        mx = fmaxf(mx, __shfl_xor(mx, off, 32));
      const float mnew = fmaxf(mrow[r], mx);
      const float corr = __expf(mrow[r] - mnew);
      mrow[r] = mnew;
      float rs = 0.0f;
#pragma unroll
      for (int nt = 0; nt < 4; ++nt) {
        const float p = __expf(st[nt][r] - mnew);
        st[nt][r] = p;
        rs += p;
      }
#pragma unroll
      for (int off = 1; off <= 8; off <<= 1)
        rs += __shfl_xor(rs, off, 32);
      lrow[r] = lrow[r] * corr + rs;
#pragma unroll
      for (int nt = 0; nt < 4; ++nt) o[nt][r] *= corr;
    }

    // P: C-layout -> per-wave LDS -> A-fragments; O += P @ V
#pragma unroll
    for (int nt = 0; nt < 4; ++nt)
#pragma unroll
      for (int r = 0; r < 8; ++r)
        Ps[wv][r + 8 * hi][nt * 16 + lo] = st[nt][r];
    // wave-private region: same-wave LDS RAW ordered by DScnt waits

#pragma unroll
    for (int ks = 0; ks < 16; ++ks) {
      const int k0 = ks * 4 + 2 * hi;
      v2f pa = *(const v2f*)&Ps[wv][lo][k0];   // adjacent -> b64
#pragma unroll
      for (int nt = 0; nt < 4; ++nt) {
        v2f vb;                                 // rows k0,k0+1 -> ds_load_2addr
        vb.x = Vs[k0][nt * 16 + lo];
        vb.y = Vs[k0 + 1][nt * 16 + lo];
        o[nt] = WMMA_F32(pa, vb, o[nt]);
      }
    }
    __syncthreads();  // protect Ks/Vs before next stage
  }

  // epilogue: O row m = q0 + r + 8*hi, normalized by row sum
#pragma unroll
  for (int r = 0; r < 8; ++r) {
    const float inv = 1.0f / lrow[r];
    const int   m   = q0 + r + 8 * hi;
#pragma unroll
    for (int nt = 0; nt < 4; ++nt)
      O[(size_t)m * kDH + nt * 16 + lo] = o[nt][r] * inv;
  }
}

// ===========================================================================
// Launcher: 3 projection GEMMs -> fused attention -> output GEMM.
// Scratch in d_ws: Q | K | V | AO, each 4096*1024 fp32 (16 MB) = 64 MB total.
// ===========================================================================
extern "C" void kernel_launch(void* const* d_in, const int* in_sizes, int n_in,
                              void* d_out, int out_size, void* d_ws, size_t ws_size,
                              hipStream_t stream) {
  const float* q    = (const float*)d_in[0];
  const float* k    = (const float*)d_in[1];
  const float* v    = (const float*)d_in[2];
  const float* mask = (const float*)d_in[3];
  const float* wq_w = (const float*)d_in[4];
  const float* wq_b = (const float*)d_in[5];
  const float* wk_w = (const float*)d_in[6];
  const float* wk_b = (const float*)d_in[7];
  const float* wv_w = (const float*)d_in[8];
  const float* wv_b = (const float*)d_in[9];
  const float* wo_w = (const float*)d_in[10];
  const float* wo_b = (const float*)d_in[11];

  const int    M   = kB * kS;            // 4096
  const size_t mat = (size_t)M * kD;     // 4096*1024 elements
  float* Qb = (float*)d_ws;
  float* Kb = Qb + mat;
  float* Vb = Kb + mat;
  float* AO = Vb + mat;

  const dim3 gGrid(kD / GBN, M / GBM);   // (8, 32)
  gemm_bias_wmma<<<gGrid, 256, 0, stream>>>(q, wq_w, wq_b, Qb, M, kD, kD);
  gemm_bias_wmma<<<gGrid, 256, 0, stream>>>(k, wk_w, wk_b, Kb, M, kD, kD);
  gemm_bias_wmma<<<gGrid, 256, 0, stream>>>(v, wv_w, wv_b, Vb, M, kD, kD);

  const dim3 aGrid(kS / 64, kH, kB);     // (32, 16, 2)
  flash_attn_wmma<<<aGrid, 128, 0, stream>>>(Qb, Kb, Vb, mask, AO);

  gemm_bias_wmma<<<gGrid, 256, 0, stream>>>(AO, wo_w, wo_b, (float*)d_out, M, kD, kD);
}